// Wrod2vec_82274393522439
// MI455X (gfx1250) — compile-verified
//
#include <hip/hip_runtime.h>

typedef __attribute__((ext_vector_type(2))) float v2f;
typedef __attribute__((ext_vector_type(4))) float v4f;
typedef __attribute__((ext_vector_type(8))) float v8f;

#define WAVES_PER_BLOCK 8
#define ROWS_PER_BLOCK  (WAVES_PER_BLOCK * 16)

// -log_sigmoid(z) = softplus(-z); stable softplus
__device__ __forceinline__ float softplusf(float x) {
    return fmaxf(x, 0.0f) + __logf(1.0f + __expf(-fabsf(x)));
}

// constant-index select tree (pure v_cndmask, avoids scratch for dynamic vector index)
__device__ __forceinline__ float vsel8(v8f v, int r) {
    float lo01 = (r & 1) ? v[1] : v[0];
    float lo23 = (r & 1) ? v[3] : v[2];
    float hi01 = (r & 1) ? v[5] : v[4];
    float hi23 = (r & 1) ? v[7] : v[6];
    float lo = (r & 2) ? lo23 : lo01;
    float hi = (r & 2) ? hi23 : hi01;
    return (r & 4) ? hi : lo;
}

__global__ __launch_bounds__(256) void w2v_loss_kernel(
    const int* __restrict__ target, const int* __restrict__ pos,
    const int* __restrict__ neg, const float* __restrict__ w1,
    const float* __restrict__ w2, float* __restrict__ partials)
{
    __shared__ float s_wave[WAVES_PER_BLOCK];

    const int lane        = threadIdx.x & 31;
    const int waveInBlock = threadIdx.x >> 5;
    const int rowBase     = (blockIdx.x * WAVES_PER_BLOCK + waveInBlock) * 16;
    const int m           = lane & 15;   // matrix row handled by this lane
    const int half        = lane >> 4;   // which 32-column half this lane loads
    const int row         = rowBase + m;

    const int t  = target[row];
    const int p  = pos[row];
    const int n0 = neg[2 * row + 0];
    const int n1 = neg[2 * row + 1];

    // each lane reads a contiguous 128B half-row (8 x b128)
    const float* hrow = w1 + (size_t)t  * 64 + 32 * half;
    const float* prow = w2 + (size_t)p  * 64 + 32 * half;
    const float* g0   = w2 + (size_t)n0 * 64 + 32 * half;
    const float* g1   = w2 + (size_t)n1 * 64 + 32 * half;

    v8f accP = {};   // diag -> dot(hidden, w2[pos])
    v8f accN = {};   // diag -> dot(hidden, w2[neg0]+w2[neg1])

#pragma unroll
    for (int c2 = 0; c2 < 8; ++c2) {
        v4f a4 = *(const v4f*)(hrow + 4 * c2);
        v4f p4 = *(const v4f*)(prow + 4 * c2);
        v4f q0 = *(const v4f*)(g0   + 4 * c2);
        v4f q1 = *(const v4f*)(g1   + 4 * c2);
        v4f n4 = q0 + q1;

        v2f aLo = {a4.x, a4.y}, aHi = {a4.z, a4.w};
        v2f pLo = {p4.x, p4.y}, pHi = {p4.z, p4.w};
        v2f nLo = {n4.x, n4.y}, nHi = {n4.z, n4.w};

        // D = A(16x4) x B(4x16) + C ; K-slots {2*half, 2*half+1} carry this
        // lane's data columns — consistent A/B mapping, so diag is the dot.
        accP = __builtin_amdgcn_wmma_f32_16x16x4_f32(false, aLo, false, pLo, (short)0, accP, false, false);
        accN = __builtin_amdgcn_wmma_f32_16x16x4_f32(false, aLo, false, nLo, (short)0, accN, false, false);
        accP = __builtin_amdgcn_wmma_f32_16x16x4_f32(false, aHi, false, pHi, (short)0, accP, false, false);
        accN = __builtin_amdgcn_wmma_f32_16x16x4_f32(false, aHi, false, nHi, (short)0, accN, false, false);
    }

    // diagonal extraction per documented 16x16 f32 C/D layout:
    // lanes 0-7 hold diag[m]=c[lane]; lanes 24-31 hold diag[m=lane-16]=c[lane-24]
    int r = (lane < 8) ? lane : ((lane >= 24) ? (lane - 24) : -1);
    float loss = 0.0f;
    if (r >= 0) {
        float pd = vsel8(accP, r);          // pos_score
        float nd = vsel8(accN, r);          // -neg_score
        loss = softplusf(-pd) + softplusf(nd);
    }

    // wave32 butterfly reduction
#pragma unroll
    for (int o = 16; o > 0; o >>= 1)
        loss += __shfl_xor(loss, o, 32);

    if (lane == 0) s_wave[waveInBlock] = loss;
    __syncthreads();
    if (threadIdx.x == 0) {
        float s = 0.0f;
#pragma unroll
        for (int i = 0; i < WAVES_PER_BLOCK; ++i) s += s_wave[i];
        partials[blockIdx.x] = s;
    }
}

__global__ __launch_bounds__(512) void w2v_reduce_kernel(
    const float* __restrict__ partials, int n, float* __restrict__ out, float scale)
{
    __shared__ float s[512];
    const int t = threadIdx.x;
    float v = 0.0f;
    for (int i = t; i < n; i += 512) v += partials[i];   // fixed order: deterministic
    s[t] = v;
    __syncthreads();
    for (int o = 256; o > 0; o >>= 1) {
        if (t < o) s[t] += s[t + o];
        __syncthreads();
    }
    if (t == 0) out[0] = s[0] * scale;
}

extern "C" void kernel_launch(void* const* d_in, const int* in_sizes, int n_in,
                              void* d_out, int out_size, void* d_ws, size_t ws_size,
                              hipStream_t stream) {
    const int*   target = (const int*)d_in[0];
    const int*   pos    = (const int*)d_in[1];
    const int*   neg    = (const int*)d_in[2];
    const float* w1     = (const float*)d_in[3];
    const float* w2     = (const float*)d_in[4];
    float*       out    = (float*)d_out;
    float*       part   = (float*)d_ws;

    const int B       = in_sizes[0];           // 65536
    const int nblocks = B / ROWS_PER_BLOCK;    // 512

    w2v_loss_kernel<<<nblocks, 256, 0, stream>>>(target, pos, neg, w1, w2, part);
    w2v_reduce_kernel<<<1, 512, 0, stream>>>(part, nblocks, out, 1.0f / (float)B);
}